// Dynamic_Person_Inference_18889266168339
// MI455X (gfx1250) — compile-verified
//
#include <hip/hip_runtime.h>
#include <hip/hip_bf16.h>

// Problem constants (from reference setup_inputs)
#define BN 8
#define TT 20
#define NNW 16
#define CC 1024
#define K2C 9
#define OO 18                 // 2*K2 conv output channels
#define MTOT (BN*TT*NNW)      // 2560 spatial positions
#define DYN_SIZE ((size_t)MTOT*CC)         // 2,621,440 floats (first output)
#define FT_SIZE  ((size_t)MTOT*K2C*CC)     // 23,592,960 floats (second output)

typedef float v2f __attribute__((ext_vector_type(2)));
typedef float v4f __attribute__((ext_vector_type(4)));
typedef float v8f __attribute__((ext_vector_type(8)));

// ---------------------------------------------------------------------------
// Stage A: dilated 3x3 offset conv as f32 WMMA GEMM.
//   out[m][o] = sum_{tap,c} pf[b, t+(kh-1)*r, n+(kw-1)*r, c] * w[o,c,kh,kw] + bias[o]
// One wave (32 threads) per 16x16 tile. grid = (160 Mtiles, 2 Otiles, 2 ratios).
// Within an M-tile the 16 rows share (b,t); n = lane%16 (N==16).
// ---------------------------------------------------------------------------
__global__ void offset_conv_wmma(const float* __restrict__ pf,
                                 const float* __restrict__ w1, const float* __restrict__ b1,
                                 const float* __restrict__ w2, const float* __restrict__ b2,
                                 float* __restrict__ off_out /* [2][MTOT*OO] */)
{
#if defined(__gfx1250__)
    const int mtile = blockIdx.x;          // 0..159
    const int otile = blockIdx.y;          // 0..1
    const int ratio = (int)blockIdx.z + 1; // 1 or 2
    const float* __restrict__ w    = (ratio == 1) ? w1 : w2;
    const float* __restrict__ bias = (ratio == 1) ? b1 : b2;
    float* __restrict__ outp = off_out + (size_t)(ratio - 1) * (MTOT * OO);

    const int lane = threadIdx.x & 31;
    const int lm   = lane & 15;
    const int half = lane >> 4;

    // A-frag row for this lane: m = mtile*16 + lm  ->  (b, t) uniform, n = lm
    const int t = mtile % TT;
    const int b = mtile / TT;
    const int n = lm;

    const int o = otile * 16 + lm;         // B-frag column = output channel
    const bool ovalid = (o < OO);

    v8f acc = {};
    for (int tap = 0; tap < K2C; ++tap) {
        const int kh = tap / 3, kw = tap % 3;
        const int ti = t + (kh - 1) * ratio;
        const int ni = n + (kw - 1) * ratio;
        const bool inb = (ti >= 0) && (ti < TT) && (ni >= 0) && (ni < NNW);
        const float* __restrict__ arow = pf + (((size_t)(b * TT + ti) * NNW + ni) * CC);
        // weight "column": w[o, c, kh, kw], stride over c is 9 floats
        const float* __restrict__ wcol = w + (size_t)(o < OO ? o : 0) * (CC * 9) + (size_t)kh * 3 + kw;

        for (int c0 = 0; c0 < CC; c0 += 4) {
            const int k = c0 + half * 2;   // this lane's two K slots
            v2f a = {0.f, 0.f};
            if (inb) a = *(const v2f*)(arow + k);      // global_load_b64
            v2f bb = {0.f, 0.f};
            if (ovalid) { bb.x = wcol[(size_t)k * 9]; bb.y = wcol[(size_t)(k + 1) * 9]; }
            acc = __builtin_amdgcn_wmma_f32_16x16x4_f32(
                false, a, false, bb, (short)0, acc, false, false);
        }
    }

    // D layout: VGPR r -> M = r + 8*half, N = lm (== channel o's column)
    if (ovalid) {
        const float bv = bias[o];
#pragma unroll
        for (int r = 0; r < 8; ++r) {
            const int mrow = mtile * 16 + r + 8 * half;
            outp[(size_t)mrow * OO + o] = acc[r] + bv;
        }
    }
#endif
}

// ---------------------------------------------------------------------------
// Stage B: deformable bilinear gather per ratio.
// One block per (b,t,n); 256 threads x float4 over C=1024.
// ft_infer (ratio 2) streams to d_out's ft_mad region with NON-TEMPORAL
// stores (94 MB write-once -> don't rinse L2; inputs + dyn_pre stay hot).
// ---------------------------------------------------------------------------
__device__ __forceinline__ v4f fetch_corner(const float* __restrict__ pf,
                                            int b, int it, int in,
                                            int pt, int pl, int c4)
{
    v4f v = {0.f, 0.f, 0.f, 0.f};
    if (it >= pt && it < pt + TT && in >= pl && in < pl + NNW) {
        const float* p = pf + (((size_t)(b * TT + (it - pt)) * NNW + (in - pl)) * CC + c4);
        v = *(const v4f*)p;
    }
    return v;
}

__global__ void deform_gather(const float* __restrict__ pf,
                              const float* __restrict__ off,   // [MTOT*OO] for this ratio
                              float* __restrict__ dyn_pre,     // [MTOT*CC] workspace
                              float* __restrict__ ft_mad,      // d_out+DYN_SIZE (ratio 2) or null
                              int ratio)
{
    const int pos = blockIdx.x;            // 0..MTOT-1
    const int n = pos & 15;
    const int t = (pos >> 4) % TT;
    const int b = pos / (TT * NNW);
    const int pt = ratio, pl = ratio;
    const int Tp = TT + 2 * pt, Np = NNW + 2 * pl;
    const int c4 = (int)threadIdx.x * 4;   // 256 threads cover C=1024

    const float* __restrict__ offp = off + (size_t)pos * OO;

    v4f acc = {0.f, 0.f, 0.f, 0.f};
    for (int k = 0; k < K2C; ++k) {
        const int kh = k / 3, kw = k % 3;
        float pos_t = (float)(t + pt) + (float)((kh - 1) * ratio) + offp[k];
        float pos_n = (float)(n + pl) + (float)((kw - 1) * ratio) + offp[K2C + k];

        const float fl_t = floorf(pos_t), fl_n = floorf(pos_n);
        const float lt_t = fminf(fmaxf(fl_t,        0.f), (float)(Tp - 1));
        const float lt_n = fminf(fmaxf(fl_n,        0.f), (float)(Np - 1));
        const float rb_t = fminf(fmaxf(fl_t + 1.f,  0.f), (float)(Tp - 1));
        const float rb_n = fminf(fmaxf(fl_n + 1.f,  0.f), (float)(Np - 1));
        pos_t = fminf(fmaxf(pos_t, 0.f), (float)(Tp - 1));
        pos_n = fminf(fmaxf(pos_n, 0.f), (float)(Np - 1));

        const float wlt_t = 1.f - fabsf(pos_t - lt_t);
        const float wrb_t = 1.f - fabsf(pos_t - rb_t);
        const float wlt_n = 1.f - fabsf(pos_n - lt_n);
        const float wrb_n = 1.f - fabsf(pos_n - rb_n);
        const float coe_lt = wlt_t * wlt_n;
        const float coe_rb = wrb_t * wrb_n;
        const float coe_lb = wrb_t * wlt_n;   // (rb_t, lt_n)
        const float coe_rt = wlt_t * wrb_n;   // (lt_t, rb_n)

        const int ilt_t = (int)lt_t, ilt_n = (int)lt_n;
        const int irb_t = (int)rb_t, irb_n = (int)rb_n;

        const v4f f_lt = fetch_corner(pf, b, ilt_t, ilt_n, pt, pl, c4);
        const v4f f_rb = fetch_corner(pf, b, irb_t, irb_n, pt, pl, c4);
        const v4f f_lb = fetch_corner(pf, b, irb_t, ilt_n, pt, pl, c4);
        const v4f f_rt = fetch_corner(pf, b, ilt_t, irb_n, pt, pl, c4);

        const v4f v = coe_lt * f_lt + coe_rb * f_rb + coe_lb * f_lb + coe_rt * f_rt;
        acc += v;

        if (ft_mad) {
            // write-once stream: non-temporal store, keep L2 for resident data
            v4f* dst = (v4f*)(ft_mad + (((size_t)pos * K2C + k) * CC + c4));
            __builtin_nontemporal_store(v, dst);
        }
    }

    const v4f ft = acc * (1.f / 9.f);
    v4f* dp = (v4f*)(dyn_pre + ((size_t)pos * CC + c4));
    if (ratio == 1) {
        *dp = ft;                              // re-read by ratio-2 pass: regular store
    } else {
        const v4f prev = *dp;                  // mean over the two ratios
        *dp = (prev + ft) * 0.5f;
    }
}

// ---------------------------------------------------------------------------
// Stage C: dyn = dyn_pre @ W_hidden.T via f32 WMMA 16x16x4, LDS-staged A.
// One block (8 waves) owns an M-tile; the 16x1024 A-tile (64 KB of the WGP's
// 320 KB LDS) is loaded once and serves 8 N-tiles -> 8x reuse. LDS row pitch
// padded to 1028 floats so same-k reads across the 16 rows hit 16 banks.
// B rows stream from L2 with prefetch ahead.
// ---------------------------------------------------------------------------
#define LDA 1028

__global__ __launch_bounds__(256) void linear_wmma(const float* __restrict__ dyn_pre,
                                                   const float* __restrict__ Wh,  // [CC,CC]
                                                   float* __restrict__ out)       // [MTOT,CC]
{
#if defined(__gfx1250__)
    __shared__ float As[16 * LDA];

    const int mtile = blockIdx.x >> 3;     // 0..159
    const int ngrp  = blockIdx.x & 7;      // 8 groups x 8 ntiles = 64 ntiles
    const int tid   = threadIdx.x;

    // Cooperative stage of A tile: row i, cols tid*4..tid*4+3
    const float* __restrict__ Abase = dyn_pre + (size_t)mtile * 16 * CC;
#pragma unroll
    for (int i = 0; i < 16; ++i) {
        const v4f val = *(const v4f*)(Abase + (size_t)i * CC + tid * 4);
        *(v4f*)(As + i * LDA + tid * 4) = val;
    }
    __syncthreads();

    const int wv    = tid >> 5;
    const int ntile = ngrp * 8 + wv;
    const int lane  = tid & 31;
    const int lm    = lane & 15;
    const int half  = lane >> 4;

    const float* __restrict__ brow  = Wh + (size_t)(ntile * 16 + lm) * CC;
    const float* __restrict__ arowL = As + lm * LDA;

    v8f acc = {};
    for (int c0 = 0; c0 < CC; c0 += 64) {
        __builtin_prefetch(brow + c0 + 256, 0, 0);   // global_prefetch_b8 ahead of stream
#pragma unroll
        for (int cc = 0; cc < 64; cc += 4) {
            const int k = c0 + cc + half * 2;
            const v2f a  = *(const v2f*)(arowL + k);   // ds_load_b64 (bank-conflict-free)
            const v2f bb = *(const v2f*)(brow + k);    // global_load_b64
            acc = __builtin_amdgcn_wmma_f32_16x16x4_f32(
                false, a, false, bb, (short)0, acc, false, false);
        }
    }

    // D layout: VGPR r -> M = r + 8*half, N = lm. Output is write-once -> NT.
#pragma unroll
    for (int r = 0; r < 8; ++r) {
        const int mrow = mtile * 16 + r + 8 * half;
        __builtin_nontemporal_store(acc[r], &out[(size_t)mrow * CC + ntile * 16 + lm]);
    }
#endif
}

// ---------------------------------------------------------------------------
// Launch: offsets (WMMA conv) -> gather r=1 -> gather r=2 (+ft_mad) -> linear.
// Workspace: [off_r1 | off_r2 | dyn_pre] floats (~10.9 MB).
// ---------------------------------------------------------------------------
extern "C" void kernel_launch(void* const* d_in, const int* in_sizes, int n_in,
                              void* d_out, int out_size, void* d_ws, size_t ws_size,
                              hipStream_t stream) {
    const float* pf = (const float*)d_in[0];   // person_features [B,T,N,C]
    const float* Wh = (const float*)d_in[1];   // W_hidden [C,C]
    const float* w1 = (const float*)d_in[2];   // p_w1 [18,C,3,3]
    const float* b1 = (const float*)d_in[3];   // p_b1 [18]
    const float* w2 = (const float*)d_in[4];   // p_w2
    const float* b2 = (const float*)d_in[5];   // p_b2

    float* out = (float*)d_out;                // [dyn | ft_mad]
    float* ws  = (float*)d_ws;
    float* off_all = ws;                            // 2 * MTOT*OO = 92,160 floats
    float* dyn_pre = ws + (size_t)2 * MTOT * OO;    // MTOT*CC floats

    // Stage A: both ratios' offset convs (WMMA f32)
    offset_conv_wmma<<<dim3(160, 2, 2), 32, 0, stream>>>(pf, w1, b1, w2, b2, off_all);

    // Stage B: deformable gathers (ratio 2 also writes ft_mad output)
    deform_gather<<<MTOT, 256, 0, stream>>>(pf, off_all, dyn_pre,
                                            nullptr, 1);
    deform_gather<<<MTOT, 256, 0, stream>>>(pf, off_all + (size_t)MTOT * OO, dyn_pre,
                                            out + DYN_SIZE, 2);

    // Stage C: final linear (WMMA f32), 160 Mtiles x 8 ntile-groups
    linear_wmma<<<160 * 8, 256, 0, stream>>>(dyn_pre, Wh, out);
}